// NormalizedCrossCorrelationLoss_57939108823826
// MI455X (gfx1250) — compile-verified
//
#include <hip/hip_runtime.h>

// ---------------------------------------------------------------------------
// Fused 9x9 local normalized-cross-correlation loss for 16x1x512x512 fp32.
// One pass over HBM (32 MB total -> ~1.4 us floor at 23.3 TB/s; halo re-reads
// hit the 192 MB L2 since the whole working set is resident).
// Vertical 9-tap box sums: LDS running sums.
// Horizontal 9-tap box sums: V_WMMA_F32_16X16X4_F32 against a banded 0/1
// matrix (exact in fp32) -- 6 K-chunks of 4 per 16x16 subtile per quantity.
// Interior tiles load via 16B transactions, asynchronously to LDS
// (GLOBAL_LOAD_ASYNC_TO_LDS_B128 / ASYNCcnt) when the toolchain exposes it.
// ---------------------------------------------------------------------------

#define WIN 9
#define PADR 4
#define TW 64                 // output tile width per workgroup
#define TH2 32                // output tile height per workgroup
#define RW (TW + 2 * PADR)    // 72: haloed region width (18 float4s)
#define RH (TH2 + 2 * PADR)   // 40: haloed region height
#define VSTR 76               // LDS row stride: multiple of 4 -> 16B-aligned
                              // rows for B128; 12-bank row step -> the 16-row
                              // WMMA A-fragment read is bank-conflict-free
#define IMG 512
#define BATCH 16
#define NTHREADS 256
#define NWAVES 8

typedef float v2f __attribute__((ext_vector_type(2)));
typedef float v8f __attribute__((ext_vector_type(8)));
typedef int   v4i __attribute__((ext_vector_type(4)));

#define GAS __attribute__((address_space(1)))
#define LAS __attribute__((address_space(3)))

#if defined(__has_builtin)
#if __has_builtin(__builtin_amdgcn_global_load_async_to_lds_b128)
#define HAVE_ASYNC_LDS 1
#endif
#if __has_builtin(__builtin_amdgcn_s_wait_asynccnt)
#define HAVE_WAIT_ASYNC 1
#endif
#endif
#ifndef HAVE_ASYNC_LDS
#define HAVE_ASYNC_LDS 0
#endif
#ifndef HAVE_WAIT_ASYNC
#define HAVE_WAIT_ASYNC 0
#endif

__global__ __launch_bounds__(NTHREADS) void ncc_tile_kernel(
    const float* __restrict__ I, const float* __restrict__ J,
    float* __restrict__ out, int mode) {
  __shared__ float sI[RH * VSTR];
  __shared__ float sJ[RH * VSTR];
  __shared__ float V[TH2 * VSTR];
  __shared__ float wsum[NWAVES];

  const int tid  = threadIdx.x;
  const int lane = tid & 31;
  const int wave = tid >> 5;

  const int tileX = blockIdx.x;  // 0..7
  const int tileY = blockIdx.y;  // 0..15
  const int b     = blockIdx.z;  // 0..15

  const long long imgBase = (long long)b * IMG * IMG;
  const int gx0 = tileX * TW - PADR;
  const int gy0 = tileY * TH2 - PADR;

  // ---- Phase 1: haloed region of I and J -> LDS (zero padding at borders)
  const bool interior =
      (gx0 >= 0) && (gy0 >= 0) && (gx0 + RW <= IMG) && (gy0 + RH <= IMG);
  if (interior) {
    // 16B-aligned fast path: 40 rows x 18 float4 per image.
    const int nvec = RH * (RW / 4);  // 720
    for (int idx = tid; idx < nvec; idx += NTHREADS) {
      const int r  = idx / (RW / 4);
      const int c  = (idx % (RW / 4)) * 4;
      const long long g = imgBase + (long long)(gy0 + r) * IMG + (gx0 + c);
      const int lo = r * VSTR + c;
#if HAVE_ASYNC_LDS
      __builtin_amdgcn_global_load_async_to_lds_b128(
          (GAS v4i*)(I + g), (LAS v4i*)&sI[lo], 0, 0);
      __builtin_amdgcn_global_load_async_to_lds_b128(
          (GAS v4i*)(J + g), (LAS v4i*)&sJ[lo], 0, 0);
#else
      *(float4*)&sI[lo] = *(const float4*)(I + g);
      *(float4*)&sJ[lo] = *(const float4*)(J + g);
#endif
    }
#if HAVE_ASYNC_LDS
#if HAVE_WAIT_ASYNC
    __builtin_amdgcn_s_wait_asynccnt(0);
#else
    asm volatile("s_wait_asynccnt 0x0" ::: "memory");
#endif
#endif
  } else {
    // Border tiles: per-element predicated loads with zero fill.
    for (int idx = tid; idx < RH * RW; idx += NTHREADS) {
      const int r = idx / RW, c = idx % RW;
      const int gy = gy0 + r, gx = gx0 + c;
      const bool in = (gy >= 0) & (gy < IMG) & (gx >= 0) & (gx < IMG);
      const long long g = imgBase + (long long)gy * IMG + gx;
      sI[r * VSTR + c] = in ? I[g] : 0.0f;
      sJ[r * VSTR + c] = in ? J[g] : 0.0f;
    }
  }

  // ---- Banded ones B-matrix fragments (24x16 band, six 4x16 K-chunks).
  // B[k][n] = 1 iff n <= k <= n+8.  Wave32 4x16 fp32 B layout:
  // lane L, vgpr j  <->  B[j + 2*(L>=16)][L%16].
  const int n  = lane & 15;
  const int hi = lane >> 4;
  v2f bfrag[6];
#pragma unroll
  for (int ci = 0; ci < 6; ++ci) {
#pragma unroll
    for (int jj = 0; jj < 2; ++jj) {
      const int k = ci * 4 + jj + 2 * hi;
      bfrag[ci][jj] = (k >= n && k <= n + 8) ? 1.0f : 0.0f;
    }
  }

  // Each wave owns one 16x16 subtile of the 32x64 tile.
  const int r0 = (wave >> 2) * 16;  // 0 or 16
  const int c0 = (wave & 3) * 16;   // 0,16,32,48

  v8f accs[5];
#pragma unroll
  for (int q = 0; q < 5; ++q) accs[q] = (v8f){};

  __syncthreads();

#pragma unroll
  for (int q = 0; q < 5; ++q) {
    // ---- Phase 2: vertical 9-tap box sum of quantity q into V (TH2 x RW)
    for (int idx = tid; idx < TH2 * RW; idx += NTHREADS) {
      const int r = idx / RW, c = idx % RW;
      float acc = 0.0f;
#pragma unroll
      for (int d = 0; d < WIN; ++d) {
        const float iv = sI[(r + d) * VSTR + c];
        const float jv = sJ[(r + d) * VSTR + c];
        const float p = (q == 0)   ? iv
                        : (q == 1) ? jv
                        : (q == 2) ? iv * iv
                        : (q == 3) ? jv * jv
                                   : iv * jv;
        acc += p;
      }
      V[r * VSTR + c] = acc;
    }
    __syncthreads();

    // ---- Phase 3: horizontal 9-tap box sum as fp32 WMMA banded matmul.
    // A (16x4 fp32) layout: lane L, vgpr j <-> A[L%16][j + 2*(L>=16)].
#pragma unroll
    for (int ci = 0; ci < 6; ++ci) {
      const int row = r0 + n;
      const int col = c0 + ci * 4 + 2 * hi;  // even -> 8B aligned
      const v2f a = *(const v2f*)&V[row * VSTR + col];
      accs[q] = __builtin_amdgcn_wmma_f32_16x16x4_f32(
          /*neg_a=*/false, a, /*neg_b=*/false, bfrag[ci],
          /*c_mod=*/(short)0, accs[q], /*reuse_a=*/false, /*reuse_b=*/false);
    }
    __syncthreads();
  }

  // ---- Epilogue: per-pixel NCC on the 8 accumulator elements per lane.
  const float area = 81.0f;
  float ccsum = 0.0f;
#pragma unroll
  for (int e = 0; e < 8; ++e) {
    const float Is = accs[0][e], Js = accs[1][e];
    const float II = accs[2][e], JJ = accs[3][e], IJ = accs[4][e];
    const float uI = Is / area, uJ = Js / area;
    const float cross = IJ - uJ * Is - uI * Js + uI * uJ * area;
    const float Ivar = II - 2.0f * uI * Is + uI * uI * area;
    const float Jvar = JJ - 2.0f * uJ * Js + uJ * uJ * area;
    ccsum += cross * cross / (Ivar * Jvar + 1e-5f);
  }

  // wave32 reduction
#pragma unroll
  for (int off = 16; off > 0; off >>= 1) ccsum += __shfl_xor(ccsum, off, 32);
  if (lane == 0) wsum[wave] = ccsum;
  __syncthreads();

  if (tid == 0) {
    float s = 0.0f;
#pragma unroll
    for (int w = 0; w < NWAVES; ++w) s += wsum[w];
    if (mode == 0) {
      atomicAdd(out, s * (-1.0f / (float)((long long)BATCH * IMG * IMG)));
    } else {
      const int bid =
          blockIdx.x + gridDim.x * (blockIdx.y + gridDim.y * blockIdx.z);
      out[bid] = s;
    }
  }
}

__global__ void zero_out_kernel(float* out) {
  if (threadIdx.x == 0) out[0] = 0.0f;
}

__global__ __launch_bounds__(256) void reduce_partials_kernel(
    const float* __restrict__ part, float* __restrict__ out, int nparts) {
  __shared__ float sm[256];
  float s = 0.0f;
  for (int i = threadIdx.x; i < nparts; i += 256) s += part[i];
  sm[threadIdx.x] = s;
  __syncthreads();
  for (int o = 128; o > 0; o >>= 1) {
    if ((int)threadIdx.x < o) sm[threadIdx.x] += sm[threadIdx.x + o];
    __syncthreads();
  }
  if (threadIdx.x == 0)
    out[0] = -sm[0] / (float)((long long)BATCH * IMG * IMG);
}

extern "C" void kernel_launch(void* const* d_in, const int* in_sizes, int n_in,
                              void* d_out, int out_size, void* d_ws,
                              size_t ws_size, hipStream_t stream) {
  (void)in_sizes; (void)n_in; (void)out_size;
  const float* I = (const float*)d_in[0];
  const float* J = (const float*)d_in[1];
  float* out = (float*)d_out;

  dim3 grid(IMG / TW, IMG / TH2, BATCH);  // 8 x 16 x 16 = 2048 blocks
  const int nblocks = grid.x * grid.y * grid.z;

  if (ws_size >= (size_t)nblocks * sizeof(float)) {
    // Deterministic two-stage reduction through scratch.
    float* parts = (float*)d_ws;
    ncc_tile_kernel<<<grid, NTHREADS, 0, stream>>>(I, J, parts, 1);
    reduce_partials_kernel<<<1, 256, 0, stream>>>(parts, out, nblocks);
  } else {
    // Fallback: atomic accumulation into the scalar output.
    zero_out_kernel<<<1, 32, 0, stream>>>(out);
    ncc_tile_kernel<<<grid, NTHREADS, 0, stream>>>(I, J, out, 0);
  }
}